// PointPillarScatter_57397942944143
// MI455X (gfx1250) — compile-verified
//
#include <hip/hip_runtime.h>

typedef __attribute__((ext_vector_type(2))) float v2f;
typedef __attribute__((ext_vector_type(4))) float v4f;
typedef __attribute__((ext_vector_type(8))) float v8f;

#define NVOX   40000
#define IN_C   64
#define OUT_C  64
#define XD     512
#define YD     512
#define BATCH  4

// ---------------------------------------------------------------------------
// Kernel 1: fill entire output [B][OUT_C][X][Y] with bias[o].
// 67,108,864 floats -> 16,777,216 float4 stores -> 65536 blocks * 256 threads.
// Non-temporal: output never re-read, and is larger than the 192MB L2.
// ---------------------------------------------------------------------------
__global__ void __launch_bounds__(256) fill_bias_kernel(
    float* __restrict__ out, const float* __restrict__ bias)
{
    size_t idx4 = (size_t)blockIdx.x * blockDim.x + threadIdx.x; // float4 index
    size_t base = idx4 * 4;
    // channel o = (base / (512*512)) % 64 ; base>>18 since 512*512 = 2^18
    int o = (int)((base >> 18) & (OUT_C - 1));
    float bv = bias[o];
    v4f v = { bv, bv, bv, bv };
    __builtin_nontemporal_store(v, reinterpret_cast<v4f*>(out + base));
}

// ---------------------------------------------------------------------------
// Kernel 2: D = feat(40000x64) x W^T(64x64) via V_WMMA_F32_16X16X4_F32,
// then scatter D[m][n] + bias[n] to out[b][n][x][y] for the 4 batches.
// Block = 128 threads = 4 waves; wave w handles N-tile w; grid.x = 2500 M-tiles.
// ---------------------------------------------------------------------------
__global__ void __launch_bounds__(128) wmma_gemm_scatter_kernel(
    const float* __restrict__ feat,   // [NVOX][IN_C]
    const float* __restrict__ w,      // [OUT_C][IN_C]
    const float* __restrict__ bias,   // [OUT_C]
    const int*   __restrict__ vidx,   // [NVOX][2]  (x, y)
    float*       __restrict__ out)    // [BATCH][OUT_C][XD][YD]
{
    const int lane   = threadIdx.x & 31;
    const int waveId = threadIdx.x >> 5;        // N-tile index 0..3
    const int mBase  = blockIdx.x * 16;         // voxel row tile
    const int nBase  = waveId * 16;             // output-channel tile

    const int half = lane >> 4;                 // 0: lanes 0-15, 1: lanes 16-31
    const int l15  = lane & 15;

    // A-frag source: row (mBase+l15), K offset half*2 within each K=4 step.
    const float* aPtr = feat + (size_t)(mBase + l15) * IN_C + half * 2;
    // B-frag source: B[k][n] = w[n][k]; lane l15 = column n, K offset half*2.
    const float* bPtr = w + (size_t)(nBase + l15) * IN_C + half * 2;

    v8f acc = {};
    #pragma unroll
    for (int ks = 0; ks < IN_C / 4; ++ks) {
        v2f a = *reinterpret_cast<const v2f*>(aPtr + ks * 4);
        v2f b = *reinterpret_cast<const v2f*>(bPtr + ks * 4);
        // 8 args: (neg_a, A, neg_b, B, c_mod, C, reuse_a, reuse_b)
        acc = __builtin_amdgcn_wmma_f32_16x16x4_f32(
            false, a, false, b, (short)0, acc, false, false);
    }

    // C/D layout: VGPR r holds element (M = r + half*8, N = l15) for this lane.
    const float bv = bias[nBase + l15];
    const size_t planeStride = (size_t)OUT_C * XD * YD;   // per-batch stride

    #pragma unroll
    for (int r = 0; r < 8; ++r) {
        const int m = mBase + r + half * 8;               // voxel id
        const float val = acc[r] + bv;
        const int x = vidx[2 * m + 0];
        const int y = vidx[2 * m + 1];
        if (x >= 0 && x < XD && y >= 0 && y < YD) {
            size_t p = ((size_t)(nBase + l15) * XD + x) * YD + y;
            #pragma unroll
            for (int bb = 0; bb < BATCH; ++bb)
                out[p + (size_t)bb * planeStride] = val;
        }
    }
}

// ---------------------------------------------------------------------------
// Inputs (setup_inputs order):
//   d_in[0] pillar_features f32 [40000*64]
//   d_in[1] conv_w          f32 [64*64]
//   d_in[2] conv_b          f32 [64]
//   d_in[3] voxel_indices   i32 [40000*2]
//   d_in[4] batch_size      (scalar, == 4)
// Output: f32 [4*64*512*512]
// ---------------------------------------------------------------------------
extern "C" void kernel_launch(void* const* d_in, const int* in_sizes, int n_in,
                              void* d_out, int out_size, void* d_ws, size_t ws_size,
                              hipStream_t stream)
{
    const float* feat = (const float*)d_in[0];
    const float* wmat = (const float*)d_in[1];
    const float* bias = (const float*)d_in[2];
    const int*   vidx = (const int*)d_in[3];
    float* out = (float*)d_out;

    // 1) bias fill: (4*64*512*512)/4 float4s / 256 threads = 65536 blocks
    fill_bias_kernel<<<65536, 256, 0, stream>>>(out, bias);

    // 2) WMMA GEMM + scatter: 40000/16 = 2500 M-tiles, 4 waves cover N=64
    wmma_gemm_scatter_kernel<<<2500, 128, 0, stream>>>(feat, wmat, bias, vidx, out);
}